// GCNEncoder_49546742726739
// MI455X (gfx1250) — compile-verified
//
#include <hip/hip_runtime.h>
#include <hip/hip_bf16.h>
#include <cstdint>

// ---------------------------------------------------------------------------
// GCN encoder: h = L2norm(relu(LN(GCN2(relu(LN(GCN1(x)))))))
// GCN(h) = segsum_dst( (hW)[src] * dinv[src]*dinv[dst] ) + hW*dinv^2 (self) + b
// fp32 throughout; GEMMs use V_WMMA_F32_16X16X4_F32 (CDNA5 fp32 matrix op).
// ---------------------------------------------------------------------------

#define N_NODES 50000
#define N_EDGES 800000
#define IN_CH   128
#define HID     128
#define OUT_CH  64

typedef __attribute__((ext_vector_type(2))) float v2f;
typedef __attribute__((ext_vector_type(8))) float v8f;

// ---------------- utility kernels ----------------

__global__ void fill_f32(float* __restrict__ p, float v, int n) {
    int i = blockIdx.x * blockDim.x + threadIdx.x;
    if (i < n) p[i] = v;
}

__global__ void degree_kernel(const long long* __restrict__ dst,
                              float* __restrict__ deg, int E) {
    int e = blockIdx.x * blockDim.x + threadIdx.x;
    if (e < E) atomicAdd(&deg[(int)dst[e]], 1.0f);
}

__global__ void rsqrt_kernel(float* __restrict__ deg, int n) {
    int i = blockIdx.x * blockDim.x + threadIdx.x;
    if (i < n) deg[i] = rsqrtf(deg[i]);  // deg >= 1 (self loop), never 0
}

// ---------------- WMMA fp32 GEMM: C[M,N] = A[M,K] @ B[K,N] ----------------
// One wave computes a 16(M) x 64(N) tile via 4 accumulators; K stepped by 4.
// A frag (16x4 f32): lanes 0-15 hold {K0,K1} of row M=lane; lanes 16-31 {K2,K3}.
// B frag (4x16 f32): lanes 0-15 hold {K0,K1} of col N=lane; lanes 16-31 {K2,K3}.
// D (16x16 f32): VGPR r, lane l -> row = r + 8*(l>=16), col = l%16.
__global__ void gemm_f32_wmma(const float* __restrict__ A,
                              const float* __restrict__ B,
                              float* __restrict__ C,
                              int M, int N, int K) {
    const int lane  = threadIdx.x & 31;
    const int wave  = (blockIdx.x * blockDim.x + threadIdx.x) >> 5;
    const int ntn   = N >> 6;              // 64-wide N tiles
    const int tm    = wave / ntn;
    const int tn    = wave - tm * ntn;
    if (tm * 16 >= M) return;              // uniform per wave: EXEC stays all-1
    const int m0    = tm * 16;
    const int n0    = tn * 64;
    const int lan15 = lane & 15;
    const int khalf = (lane >> 4) << 1;    // 0 for lanes 0-15, 2 for 16-31

    v8f acc[4] = {};
    const float* __restrict__ arow = A + (size_t)(m0 + lan15) * K;

    for (int k = 0; k < K; k += 4) {
        v2f a;
        a.x = arow[k + khalf];
        a.y = arow[k + khalf + 1];
#pragma unroll
        for (int t = 0; t < 4; ++t) {
            const int n = n0 + t * 16 + lan15;
            v2f b;
            b.x = B[(size_t)(k + khalf) * N + n];
            b.y = B[(size_t)(k + khalf + 1) * N + n];
            acc[t] = __builtin_amdgcn_wmma_f32_16x16x4_f32(
                false, a, false, b, (short)0, acc[t], false, false);
        }
    }

    const int rowoff = (lane >> 4) << 3;   // 0 or 8
#pragma unroll
    for (int t = 0; t < 4; ++t) {
        const int n = n0 + t * 16 + lan15;
#pragma unroll
        for (int r = 0; r < 8; ++r) {
            C[(size_t)(m0 + r + rowoff) * N + n] = acc[t][r];
        }
    }
}

// ---------------- edge scatter: Agg[dst] += H[src] * dinv[src]*dinv[dst] ----
template <int CH>
__global__ void scatter_edges(const float* __restrict__ H,
                              float* __restrict__ Agg,
                              const long long* __restrict__ src,
                              const long long* __restrict__ dst,
                              const float* __restrict__ dinv) {
    constexpr int SH = (CH == 128) ? 7 : 6;
    long long gid = (long long)blockIdx.x * blockDim.x + threadIdx.x;
    long long e   = gid >> SH;
    int c         = (int)(gid & (CH - 1));
    if (e >= N_EDGES) return;
    int s = (int)src[e];
    int d = (int)dst[e];
    float w = dinv[s] * dinv[d];
    atomicAdd(&Agg[(size_t)d * CH + c], H[(size_t)s * CH + c] * w);
}

// ---------------- wave32 reduction ----------------
__device__ __forceinline__ float wave_sum(float v) {
#pragma unroll
    for (int off = 16; off >= 1; off >>= 1) v += __shfl_xor(v, off, 32);
    return v;
}

// ---------------- LN + bias + self-loop + ReLU over 128 ch (in-place H1) ----
__global__ void ln_relu_128(float* __restrict__ H,         // hW (in), post-act (out)
                            const float* __restrict__ Agg, // edge aggregate
                            const float* __restrict__ dinv,
                            const float* __restrict__ b,
                            const float* __restrict__ g,
                            const float* __restrict__ beta) {
    int node = blockIdx.x * 8 + (threadIdx.x >> 5);
    if (node >= N_NODES) return;
    int lane = threadIdx.x & 31;
    float di = dinv[node];
    float sl = di * di;                    // self-loop norm
    size_t base = (size_t)node * 128;

    float v[4], s = 0.f, ss = 0.f;
#pragma unroll
    for (int j = 0; j < 4; ++j) {
        int c = lane + 32 * j;
        float t = Agg[base + c] + H[base + c] * sl + b[c];
        v[j] = t; s += t; ss += t * t;
    }
    s  = wave_sum(s);
    ss = wave_sum(ss);
    float mu   = s * (1.0f / 128.0f);
    float var  = ss * (1.0f / 128.0f) - mu * mu;
    float rstd = rsqrtf(var + 1e-5f);
#pragma unroll
    for (int j = 0; j < 4; ++j) {
        int c = lane + 32 * j;
        float y = (v[j] - mu) * rstd * g[c] + beta[c];
        H[base + c] = fmaxf(y, 0.0f);
    }
}

// ---------------- LN + bias + self-loop + ReLU + L2-normalize (64 ch) ------
__global__ void ln_relu_l2_64(const float* __restrict__ H,
                              const float* __restrict__ Agg,
                              const float* __restrict__ dinv,
                              const float* __restrict__ b,
                              const float* __restrict__ g,
                              const float* __restrict__ beta,
                              float* __restrict__ out) {
    int node = blockIdx.x * 8 + (threadIdx.x >> 5);
    if (node >= N_NODES) return;
    int lane = threadIdx.x & 31;
    float di = dinv[node];
    float sl = di * di;
    size_t base = (size_t)node * 64;
    int c0 = lane, c1 = lane + 32;

    float v0 = Agg[base + c0] + H[base + c0] * sl + b[c0];
    float v1 = Agg[base + c1] + H[base + c1] * sl + b[c1];
    float s  = wave_sum(v0 + v1);
    float ss = wave_sum(v0 * v0 + v1 * v1);
    float mu   = s * (1.0f / 64.0f);
    float var  = ss * (1.0f / 64.0f) - mu * mu;
    float rstd = rsqrtf(var + 1e-5f);
    float y0 = fmaxf((v0 - mu) * rstd * g[c0] + beta[c0], 0.0f);
    float y1 = fmaxf((v1 - mu) * rstd * g[c1] + beta[c1], 0.0f);
    float nsq = wave_sum(y0 * y0 + y1 * y1);
    float scale = 1.0f / fmaxf(sqrtf(nsq), 1e-12f);
    out[base + c0] = y0 * scale;
    out[base + c1] = y1 * scale;
}

// ---------------------------------------------------------------------------

extern "C" void kernel_launch(void* const* d_in, const int* in_sizes, int n_in,
                              void* d_out, int out_size, void* d_ws, size_t ws_size,
                              hipStream_t stream) {
    const float*     x     = (const float*)d_in[0];
    const long long* ei    = (const long long*)d_in[1];   // int64 [2, E]
    const float*     W1    = (const float*)d_in[2];
    const float*     b1    = (const float*)d_in[3];
    const float*     g1    = (const float*)d_in[4];
    const float*     beta1 = (const float*)d_in[5];
    const float*     W2    = (const float*)d_in[6];
    const float*     b2    = (const float*)d_in[7];
    const float*     g2    = (const float*)d_in[8];
    const float*     beta2 = (const float*)d_in[9];
    float*           out   = (float*)d_out;

    const long long* src = ei;
    const long long* dst = ei + N_EDGES;

    // workspace layout (floats): dinv | H1 | A1 | H2 | A2  = N*(1+128+128+64+64)
    float* ws   = (float*)d_ws;
    float* dinv = ws;
    float* H1   = dinv + N_NODES;
    float* A1   = H1 + (size_t)N_NODES * HID;
    float* H2   = A1 + (size_t)N_NODES * HID;
    float* A2   = H2 + (size_t)N_NODES * OUT_CH;

    // init: deg=1 (self loops), aggregates = 0
    fill_f32<<<(N_NODES + 255) / 256, 256, 0, stream>>>(dinv, 1.0f, N_NODES);
    fill_f32<<<(N_NODES * HID + 255) / 256, 256, 0, stream>>>(A1, 0.0f, N_NODES * HID);
    fill_f32<<<(N_NODES * OUT_CH + 255) / 256, 256, 0, stream>>>(A2, 0.0f, N_NODES * OUT_CH);

    // degrees -> dinv = rsqrt(deg)
    degree_kernel<<<(N_EDGES + 255) / 256, 256, 0, stream>>>(dst, dinv, N_EDGES);
    rsqrt_kernel<<<(N_NODES + 255) / 256, 256, 0, stream>>>(dinv, N_NODES);

    // layer 1: H1 = x @ W1  (50000x128x128), WMMA fp32
    {
        int waves  = (N_NODES / 16) * (HID / 64);          // 3125 * 2
        int blocks = (waves + 3) / 4;                      // 4 waves / block
        gemm_f32_wmma<<<blocks, 128, 0, stream>>>(x, W1, H1, N_NODES, HID, IN_CH);
    }
    {
        long long thr = (long long)N_EDGES * HID;
        scatter_edges<HID><<<(int)((thr + 255) / 256), 256, 0, stream>>>(H1, A1, src, dst, dinv);
    }
    ln_relu_128<<<(N_NODES + 7) / 8, 256, 0, stream>>>(H1, A1, dinv, b1, g1, beta1);

    // layer 2: H2 = H1 @ W2  (50000x128x64), WMMA fp32
    {
        int waves  = (N_NODES / 16) * (OUT_CH / 64);       // 3125
        int blocks = (waves + 3) / 4;
        gemm_f32_wmma<<<blocks, 128, 0, stream>>>(H1, W2, H2, N_NODES, OUT_CH, HID);
    }
    {
        long long thr = (long long)N_EDGES * OUT_CH;
        scatter_edges<OUT_CH><<<(int)((thr + 255) / 256), 256, 0, stream>>>(H2, A2, src, dst, dinv);
    }
    ln_relu_l2_64<<<(N_NODES + 7) / 8, 256, 0, stream>>>(H2, A2, dinv, b2, g2, beta2, out);
}